// Predict_9259949490487
// MI455X (gfx1250) — compile-verified
//
#include <hip/hip_runtime.h>
#include <stdint.h>

// YOLO post-process for MI455X (gfx1250, wave32).
// decode(+sigmoid/exp/threshold, class-major scores) -> per-(image,class) NMS with
// the score row DMA'd into LDS by the Tensor Data Mover (TENSOR_LOAD_TO_LDS,
// TENSORcnt) -> per-image bitonic top-150 with async global->LDS staging
// (global_load_async_to_lds_b128, ASYNCcnt). No matmul exists in this op, so WMMA
// is not applicable; TDM + async-LDS + wave32 shuffles are the CDNA5 paths used.
//
// Workspace (~50.3 MB of d_ws):
//   boxes  : 32*16128*4 f32  (x1,y1,x2,y2)
//   scores : 32*20*16128 f32 (thresholded: s>0.3 ? s : -1e30), layout [b][c][n]
//   kscore : 32*20*150 f32 ; kidx : 32*20*150 i32

#define NCAND   16128
#define NCLASS  20
#define NBATCH  32
#define MAXBOX  150
#define SCORE_TH 0.3f
#define IOU_TH   0.1f
#define NEGV    (-1e30f)

typedef unsigned int v4u __attribute__((ext_vector_type(4)));
typedef int          v8i __attribute__((ext_vector_type(8)));
typedef int          v4i __attribute__((ext_vector_type(4)));

__device__ __forceinline__ float sigmoidf(float x) { return 1.0f / (1.0f + expf(-x)); }

// ---------------------------------------------------------------- decode ----
__global__ __launch_bounds__(256) void k_decode(
    const float* __restrict__ fm0, const float* __restrict__ fm1,
    const float* __restrict__ fm2, const float* __restrict__ anchors,
    float* __restrict__ boxes, float* __restrict__ scores)
{
    int g = blockIdx.x * 256 + threadIdx.x;
    if (g >= NBATCH * NCAND) return;
    int b = g / NCAND, n = g % NCAND;

    const float* fm; int gw, abase, m;
    if (n < 768)        { fm = fm0; gw = 16; abase = 6; m = n; }
    else if (n < 3840)  { fm = fm1; gw = 32; abase = 3; m = n - 768; }
    else                { fm = fm2; gw = 64; abase = 0; m = n - 3840; }

    int a = m % 3, cell = m / 3, gx = cell % gw, gy = cell / gw;
    float stride = 512.0f / (float)gw;
    const float* p = fm + (((size_t)b * gw + gy) * gw + gx) * 75 + a * 25;

    // centers = (sigmoid + grid) * stride ; sizes = exp * anchor (ratios cancel)
    float cx = (sigmoidf(p[0]) + (float)gx) * stride;
    float cy = (sigmoidf(p[1]) + (float)gy) * stride;
    float w  = expf(p[2]) * anchors[(abase + a) * 2 + 0];
    float h  = expf(p[3]) * anchors[(abase + a) * 2 + 1];
    float4 bx = make_float4(cx - 0.5f * w, cy - 0.5f * h, cx + 0.5f * w, cy + 0.5f * h);
    ((float4*)boxes)[(size_t)b * NCAND + n] = bx;

    float conf = sigmoidf(p[4]);
    float* srow = scores + (size_t)b * NCLASS * NCAND + n;   // [b][c][n]
    #pragma unroll
    for (int c = 0; c < NCLASS; ++c) {
        float s = conf * sigmoidf(p[5 + c]);
        srow[(size_t)c * NCAND] = (s > SCORE_TH) ? s : NEGV;  // coalesced per class
    }
}

// ------------------------------------------------------------------- NMS ----
__global__ __launch_bounds__(1024) void k_nms(
    const float* __restrict__ boxes, const float* __restrict__ scores,
    int* __restrict__ kidx, float* __restrict__ kscore)
{
    int c = blockIdx.x % NCLASS, b = blockIdx.x / NCLASS;
    int tid = threadIdx.x;

    __shared__ __align__(16) float s_sc[NCAND];   // 64512 B
    __shared__ float s_rv[32];
    __shared__ int   s_ri[32];
    __shared__ float s_box[4];
    __shared__ int   s_pick;

    // --- TDM: DMA the whole (b,c) score row (64.5 KB) into LDS -------------
    // One wave issues a single TENSOR_LOAD_TO_LDS (D# per ISA ch.8), waits on
    // TENSORcnt, then the workgroup barrier releases everyone.
    const float* src = scores + (size_t)(b * NCLASS + c) * NCAND;
    if (tid < 32) {
        unsigned lds = (unsigned)(uintptr_t)(void*)s_sc;       // LDS byte offset
        unsigned long long ga = (unsigned long long)(uintptr_t)(const void*)src;
        v4u g0;
        g0[0] = 1u;                                            // count=1, user D#
        g0[1] = lds;                                           // lds_addr
        g0[2] = (unsigned)(ga & 0xFFFFFFFFull);                // global_addr[31:0]
        g0[3] = (unsigned)((ga >> 32) & 0x01FFFFFFull) | 0x80000000u; // [56:32] | type=2
        v8i g1;
        g1[0] = 0x20000;                    // workgroup_mask=0, data_size=2 (4B)
        g1[1] = (int)((NCAND & 0xFFFF) << 16);        // tensor_dim0[15:0] @ [63:48]
        g1[2] = (int)((NCAND >> 16) | (1 << 16));     // dim0[31:16] | tensor_dim1=1
        g1[3] = (int)((NCAND & 0xFFFF) << 16);        // dim1[31:16]=0 | tile_dim0
        g1[4] = 1;                                    // tile_dim1=1, tile_dim2=0
        g1[5] = NCAND;                                // tensor_dim0_stride[31:0]
        g1[6] = 0;                                    // stride hi, dim1_stride lo
        g1[7] = 0;
        v4i gz4  = {0, 0, 0, 0};                      // groups 2/3 unused (<=2D)
        v8i gz8  = {0, 0, 0, 0, 0, 0, 0, 0};          // extra group (clang-23 form)
        __builtin_amdgcn_tensor_load_to_lds(g0, g1, gz4, gz4, gz8, 0);
        __builtin_amdgcn_s_wait_tensorcnt(0);
    }
    __syncthreads();

    const float4* boxrow = (const float4*)boxes + (size_t)b * NCAND;
    float4* s4 = (float4*)s_sc;                      // NCAND/4 = 4032 quads
    int outb = (b * NCLASS + c) * MAXBOX;

    for (int it = 0; it < MAXBOX; ++it) {
        // ---- argmax over LDS scores, b128 reads (first-max tie-break) -----
        float bv = NEGV; int bi = 0x7FFFFFFF;
        #pragma unroll
        for (int k = 0; k < 4; ++k) {
            int t4 = tid + k * 1024;
            if (t4 < NCAND / 4) {
                float4 v = s4[t4];
                int tb = t4 * 4;
                if (v.x > bv) { bv = v.x; bi = tb;     }
                if (v.y > bv) { bv = v.y; bi = tb + 1; }
                if (v.z > bv) { bv = v.z; bi = tb + 2; }
                if (v.w > bv) { bv = v.w; bi = tb + 3; }
            }
        }
        #pragma unroll
        for (int off = 16; off; off >>= 1) {          // wave32 shuffle reduce
            float ov = __shfl_xor(bv, off, 32);
            int   oi = __shfl_xor(bi, off, 32);
            if (ov > bv || (ov == bv && oi < bi)) { bv = ov; bi = oi; }
        }
        if ((tid & 31) == 0) { s_rv[tid >> 5] = bv; s_ri[tid >> 5] = bi; }
        __syncthreads();
        if (tid < 32) {
            bv = s_rv[tid]; bi = s_ri[tid];
            #pragma unroll
            for (int off = 16; off; off >>= 1) {
                float ov = __shfl_xor(bv, off, 32);
                int   oi = __shfl_xor(bi, off, 32);
                if (ov > bv || (ov == bv && oi < bi)) { bv = ov; bi = oi; }
            }
            if (tid == 0) {
                if (bv > -1.0e29f) {
                    s_pick = bi;
                    float4 pb = boxrow[bi];
                    s_box[0] = pb.x; s_box[1] = pb.y; s_box[2] = pb.z; s_box[3] = pb.w;
                    kidx[outb + it]   = bi;
                    kscore[outb + it] = bv;
                } else {
                    s_pick = -1;
                }
            }
        }
        __syncthreads();
        if (s_pick < 0) {   // nothing left: fill remaining slots, uniform exit
            if (tid == 0) {
                for (int r = it; r < MAXBOX; ++r) { kidx[outb + r] = -1; kscore[outb + r] = -1.0f; }
            }
            break;
        }
        // ---- suppress IoU > 0.1 vs pick (pick suppresses itself) ----------
        float px1 = s_box[0], py1 = s_box[1], px2 = s_box[2], py2 = s_box[3];
        float parea = (px2 - px1) * (py2 - py1);
        #pragma unroll
        for (int k = 0; k < 4; ++k) {
            int t4 = tid + k * 1024;
            if (t4 >= NCAND / 4) break;
            float4 sv = s4[t4];
            bool a0 = sv.x > -1.0e29f, a1 = sv.y > -1.0e29f;
            bool a2 = sv.z > -1.0e29f, a3 = sv.w > -1.0e29f;
            if (!(a0 | a1 | a2 | a3)) continue;       // whole quad dead: skip
            int tb = t4 * 4;
            bool ch = false;
            #pragma unroll
            for (int e = 0; e < 4; ++e) {
                bool alive = (e == 0) ? a0 : (e == 1) ? a1 : (e == 2) ? a2 : a3;
                if (!alive) continue;
                float4 q = boxrow[tb + e];            // L2-resident boxes
                float ix1 = fmaxf(px1, q.x), iy1 = fmaxf(py1, q.y);
                float ix2 = fminf(px2, q.z), iy2 = fminf(py2, q.w);
                float inter = fmaxf(ix2 - ix1, 0.0f) * fmaxf(iy2 - iy1, 0.0f);
                float qarea = (q.z - q.x) * (q.w - q.y);
                float iou = inter / (parea + qarea - inter);
                if (iou > IOU_TH) {
                    if (e == 0) sv.x = NEGV; else if (e == 1) sv.y = NEGV;
                    else if (e == 2) sv.z = NEGV; else sv.w = NEGV;
                    ch = true;
                }
            }
            if (ch) s4[t4] = sv;                      // b128 write-back
        }
        __syncthreads();
    }
}

// --------------------------------------------------------- per-image topk ---
__global__ __launch_bounds__(1024) void k_topk(
    const float* __restrict__ boxes, const int* __restrict__ kidx,
    const float* __restrict__ kscore, float* __restrict__ out)
{
    const int M = NCLASS * MAXBOX;           // 3000
    const int NS = 4096;
    int b = blockIdx.x, tid = threadIdx.x;
    __shared__ unsigned long long keys[NS];  // 32 KB
    __shared__ __align__(16) float s_ks[M];  // 12 KB

    // async-stage the 3000 kept-scores into LDS (ASYNCcnt path)
    if (tid < M / 4) {
        unsigned lo = (unsigned)(uintptr_t)(void*)s_ks + (unsigned)tid * 16u;
        unsigned long long ga =
            (unsigned long long)(uintptr_t)(const void*)(kscore + (size_t)b * M + tid * 4);
        asm volatile("global_load_async_to_lds_b128 %0, %1, off"
                     :: "v"(lo), "v"(ga) : "memory");
    }
    asm volatile("s_wait_asynccnt 0" ::: "memory");
    __syncthreads();

    for (int t = tid; t < NS; t += 1024) {
        unsigned long long key = 0ull;       // pad: smaller than any real entry
        if (t < M) {
            unsigned bits = __float_as_uint(s_ks[t]);
            unsigned enc = (bits & 0x80000000u) ? ~bits : (bits | 0x80000000u); // order-preserving
            key = ((unsigned long long)enc << 32) | (unsigned)(0xFFFFFFFFu - (unsigned)t);
        }
        keys[t] = key;
    }
    __syncthreads();

    // bitonic sort, descending (stable wrt index via inverted-low-word)
    for (unsigned k = 2; k <= (unsigned)NS; k <<= 1) {
        for (unsigned j = k >> 1; j > 0; j >>= 1) {
            for (unsigned i = tid; i < (unsigned)NS; i += 1024) {
                unsigned ixj = i ^ j;
                if (ixj > i) {
                    unsigned long long a = keys[i], bb = keys[ixj];
                    bool sw = ((i & k) == 0) ? (a < bb) : (a > bb);
                    if (sw) { keys[i] = bb; keys[ixj] = a; }
                }
            }
            __syncthreads();
        }
    }

    if (tid < MAXBOX) {
        unsigned long long key = keys[tid];
        unsigned j = 0xFFFFFFFFu - (unsigned)(key & 0xFFFFFFFFu);
        if (j >= (unsigned)M) j = 0;                   // defensive
        float sc = s_ks[j];
        bool ok = sc > 0.0f;
        int cls = (int)(j / MAXBOX);

        float4 ob = make_float4(-1.0f, -1.0f, -1.0f, -1.0f);
        if (ok) {
            int idx = kidx[(size_t)b * M + j];
            ob = ((const float4*)boxes)[(size_t)b * NCAND + idx];
        }
        float* oboxes  = out;                               // 32*150*4
        float* oscores = out + NBATCH * MAXBOX * 4;         // 32*150
        float* olabels = oscores + NBATCH * MAXBOX;         // 32*150
        ((float4*)oboxes)[b * MAXBOX + tid] = ob;
        oscores[b * MAXBOX + tid] = ok ? sc : -1.0f;
        olabels[b * MAXBOX + tid] = ok ? (float)cls : -1.0f;
    }
}

// ---------------------------------------------------------------- launch ----
extern "C" void kernel_launch(void* const* d_in, const int* in_sizes, int n_in,
                              void* d_out, int out_size, void* d_ws, size_t ws_size,
                              hipStream_t stream)
{
    const float* fm0     = (const float*)d_in[0];
    const float* fm1     = (const float*)d_in[1];
    const float* fm2     = (const float*)d_in[2];
    const float* anchors = (const float*)d_in[3];
    // d_in[4] is CLASS==20 (compile-time constant here)

    float* boxes  = (float*)d_ws;                                  // 8.25 MB
    float* scores = boxes  + (size_t)NBATCH * NCAND * 4;           // 41.3 MB
    float* kscore = scores + (size_t)NBATCH * NCLASS * NCAND;      // 0.38 MB
    int*   kidx   = (int*)(kscore + (size_t)NBATCH * NCLASS * MAXBOX);

    k_decode<<<(NBATCH * NCAND + 255) / 256, 256, 0, stream>>>(
        fm0, fm1, fm2, anchors, boxes, scores);
    k_nms<<<NBATCH * NCLASS, 1024, 0, stream>>>(boxes, scores, kidx, kscore);
    k_topk<<<NBATCH, 1024, 0, stream>>>(boxes, kidx, kscore, (float*)d_out);
}